// GNN_3006477107412
// MI455X (gfx1250) — compile-verified
//
#include <hip/hip_runtime.h>
#include <hip/hip_bf16.h>

typedef __attribute__((ext_vector_type(2))) float v2f;
typedef __attribute__((ext_vector_type(8))) float v8f;

#define N_NODES 50000
#define N_EDGES 800000
#define NFEAT   128
#define NHID    64
#define NOUT    64
#define NCAT    192   // NHID + NHID + NOUT
#define NCLASS  8

// ---------------------------------------------------------------------------
// Degree / normalization:  deg[i] = 1 (self loop) + #edges with dst==i
//                          dis[i] = rsqrt(deg[i])   (deg >= 1 always)
// ---------------------------------------------------------------------------
__global__ void k_init_deg(float* __restrict__ deg) {
    int i = blockIdx.x * blockDim.x + threadIdx.x;
    if (i < N_NODES) deg[i] = 1.0f;
}

__global__ void k_count_deg(const int* __restrict__ ei, float* __restrict__ deg) {
    int e = blockIdx.x * blockDim.x + threadIdx.x;
    if (e < N_EDGES) atomicAdd(&deg[ei[N_EDGES + e]], 1.0f);
}

__global__ void k_finalize_dis(float* __restrict__ deg) {
    int i = blockIdx.x * blockDim.x + threadIdx.x;
    if (i < N_NODES) deg[i] = rsqrtf(deg[i]);
}

// ---------------------------------------------------------------------------
// Transpose + zero-pad B (KxN row-major) into BT (NP x K row-major), NP=ceil16(N).
// Removes all column guards (and thus EXEC churn) from the WMMA inner loop and
// makes each lane's two K-values contiguous (single global_load_b64).
// ---------------------------------------------------------------------------
__global__ void k_prep_bt(const float* __restrict__ B, float* __restrict__ BT,
                          int K, int N, int NP) {
    int idx = blockIdx.x * blockDim.x + threadIdx.x;
    if (idx >= NP * K) return;
    const int n = idx / K;
    const int k = idx - n * K;
    BT[(long)n * K + k] = (n < N) ? B[(long)k * N + n] : 0.0f;
}

// ---------------------------------------------------------------------------
// fp32 WMMA GEMM:  C[:, coff : coff+Nstore] = A(MxK, lda) @ B(KxN) [+ bias]
// B comes pre-transposed/padded as BT (NP x K). One 16x16 output tile per
// wave; branch-free inner loop: 2x global_load_b64 + v_wmma_f32_16x16x4_f32.
// All WMMA-relevant control flow is wave-uniform (EXEC all-1s requirement).
// ---------------------------------------------------------------------------
__global__ void k_gemm_wmma(const float* __restrict__ A, int lda,
                            const float* __restrict__ BT,
                            float* __restrict__ C, int ldc, int coff,
                            int M, int NP, int Nstore, int K,
                            const float* __restrict__ bias) {
    const int lane = threadIdx.x & 31;
    const int wave = threadIdx.x >> 5;
    const int tiles_n = NP >> 4;
    const int tiles_m = (M + 15) >> 4;
    const int tile = blockIdx.x * (blockDim.x >> 5) + wave;
    if (tile >= tiles_m * tiles_n) return;          // whole-wave uniform exit
    const int tm = tile / tiles_n;
    const int tn = tile % tiles_n;

    // A: lanes 0-15 hold rows M=0..15 with K-pair {0,1}; lanes 16-31 K-pair {2,3}
    int rowA = tm * 16 + (lane & 15);
    if (rowA >= M) rowA = M - 1;                    // load clamp (M%16==0 here anyway)
    const int koff = (lane >> 4) << 1;
    // B: lanes 0-15 hold cols N=0..15 with K-pair {0,1}; lanes 16-31 K-pair {2,3}
    const int colB = tn * 16 + (lane & 15);         // always < NP (padded)

    const float* __restrict__ arow = A  + (long)rowA * lda + koff;
    const float* __restrict__ brow = BT + (long)colB * K   + koff;

    v8f acc = {};
    #pragma unroll 4
    for (int k = 0; k < K; k += 4) {
        v2f a, b;
        a.x = arow[k];
        a.y = arow[k + 1];
        b.x = brow[k];
        b.y = brow[k + 1];
        acc = __builtin_amdgcn_wmma_f32_16x16x4_f32(
                  false, a, false, b, (short)0, acc, false, false);
    }

    // C/D: VGPR r -> row tm*16 + 8*(lane>=16) + r, col = lane&15
    if (colB < Nstore) {
        const float badd = bias ? bias[colB] : 0.0f;
        const int rbase = tm * 16 + ((lane >> 4) << 3);
        #pragma unroll
        for (int r = 0; r < 8; ++r) {
            const int row = rbase + r;
            if (row < M) C[(long)row * ldc + coff + colB] = acc[r] + badd;
        }
    }
}

// ---------------------------------------------------------------------------
// Self-loop + bias init of an output slice:
//   cat[i, coff+f] = h[i,f] * dis[i]^2 + b[f]
// ---------------------------------------------------------------------------
__global__ void k_self_loop_bias(const float* __restrict__ h,
                                 const float* __restrict__ dis,
                                 const float* __restrict__ b,
                                 float* __restrict__ cat, int coff) {
    int idx = blockIdx.x * blockDim.x + threadIdx.x;
    if (idx >= N_NODES * 64) return;
    const int i = idx >> 6;         // / 64
    const int f = idx & 63;         // % 64
    const float d = dis[i];
    cat[(long)i * NCAT + coff + f] = h[idx] * d * d + b[f];
}

// ---------------------------------------------------------------------------
// Edge scatter: one wave per edge, 64 features -> 2 per lane (coalesced).
//   cat[dst, coff+f] += h[src,f] * dis[src]*dis[dst]
// Working set (~51 MB) fits in the 192 MB L2, so these atomics resolve in L2.
// ---------------------------------------------------------------------------
__global__ void k_edge_agg(const int* __restrict__ ei,
                           const float* __restrict__ h,
                           const float* __restrict__ dis,
                           float* __restrict__ cat, int coff) {
    const int wid  = (blockIdx.x * blockDim.x + threadIdx.x) >> 5;
    const int lane = threadIdx.x & 31;
    if (wid >= N_EDGES) return;
    const int s = ei[wid];
    const int d = ei[N_EDGES + wid];
    const float w = dis[s] * dis[d];
    const float v0 = h[(long)s * 64 + lane]      * w;
    const float v1 = h[(long)s * 64 + lane + 32] * w;
    float* orow = cat + (long)d * NCAT + coff;
    atomicAdd(orow + lane,      v0);
    atomicAdd(orow + lane + 32, v1);
}

__global__ void k_relu_slice(float* __restrict__ cat, int coff) {
    int idx = blockIdx.x * blockDim.x + threadIdx.x;
    if (idx >= N_NODES * 64) return;
    const int i = idx >> 6;
    const int f = idx & 63;
    float* p = cat + (long)i * NCAT + coff + f;
    *p = fmaxf(*p, 0.0f);
}

// ---------------------------------------------------------------------------
// In-place row log-softmax over NCLASS=8 logits.
// ---------------------------------------------------------------------------
__global__ void k_log_softmax(float* __restrict__ out) {
    int i = blockIdx.x * blockDim.x + threadIdx.x;
    if (i >= N_NODES) return;
    float v[NCLASS];
    float m = -3.0e38f;
    #pragma unroll
    for (int c = 0; c < NCLASS; ++c) { v[c] = out[(long)i * NCLASS + c]; m = fmaxf(m, v[c]); }
    float s = 0.0f;
    #pragma unroll
    for (int c = 0; c < NCLASS; ++c) s += __expf(v[c] - m);
    const float ls = __logf(s);
    #pragma unroll
    for (int c = 0; c < NCLASS; ++c) out[(long)i * NCLASS + c] = v[c] - m - ls;
}

// ---------------------------------------------------------------------------
extern "C" void kernel_launch(void* const* d_in, const int* in_sizes, int n_in,
                              void* d_out, int out_size, void* d_ws, size_t ws_size,
                              hipStream_t stream) {
    const float* x  = (const float*)d_in[0];
    const int*   ei = (const int*)  d_in[1];
    const float* W1 = (const float*)d_in[2];
    const float* b1 = (const float*)d_in[3];
    const float* W2 = (const float*)d_in[4];
    const float* b2 = (const float*)d_in[5];
    const float* W3 = (const float*)d_in[6];
    const float* b3 = (const float*)d_in[7];
    const float* Wl = (const float*)d_in[8];
    const float* bl = (const float*)d_in[9];
    float* out = (float*)d_out;

    // Workspace layout (total ~52.3 MB):
    //   dis : N_NODES floats                 @ 0
    //   BT  : transposed/padded weights      @ 256 KB  (<= 32 KB used)
    //   h   : N_NODES x 64 floats            @ 1 MB
    //   cat : N_NODES x 192 floats           @ 1 MB + 12.8 MB
    char* ws = (char*)d_ws;
    float* dis = (float*)ws;
    float* bt  = (float*)(ws + (size_t)(256 << 10));
    float* h   = (float*)(ws + (size_t)(1 << 20));
    float* cat = (float*)(ws + (size_t)(1 << 20) + (size_t)N_NODES * 64 * 4);

    const int TPB = 256;
    const int nodeBlk = (N_NODES + TPB - 1) / TPB;
    const int edgeBlk = (N_EDGES + TPB - 1) / TPB;
    const int featBlk = (N_NODES * 64 + TPB - 1) / TPB;
    const int edgeWaveBlk = (N_EDGES * 32 + TPB - 1) / TPB;

    // deg -> dis
    k_init_deg    <<<nodeBlk, TPB, 0, stream>>>(dis);
    k_count_deg   <<<edgeBlk, TPB, 0, stream>>>(ei, dis);
    k_finalize_dis<<<nodeBlk, TPB, 0, stream>>>(dis);

    auto gemm = [&](const float* A, int lda, const float* B,
                    float* C, int ldc, int coff, int M, int N, int K,
                    const float* bias) {
        const int NP = ((N + 15) / 16) * 16;
        k_prep_bt<<<(NP * K + TPB - 1) / TPB, TPB, 0, stream>>>(B, bt, K, N, NP);
        const int tiles = ((M + 15) / 16) * (NP / 16);
        const int wavesPerBlock = TPB / 32;
        const int blocks = (tiles + wavesPerBlock - 1) / wavesPerBlock;
        k_gemm_wmma<<<blocks, TPB, 0, stream>>>(A, lda, bt, C, ldc, coff,
                                                M, NP, N, K, bias);
    };

    // Layer 1: h = x @ W1 ; cat[:,0:64] = relu(agg(h) + b1)
    gemm(x, NFEAT, W1, h, NHID, 0, N_NODES, NHID, NFEAT, nullptr);
    k_self_loop_bias<<<featBlk, TPB, 0, stream>>>(h, dis, b1, cat, 0);
    k_edge_agg      <<<edgeWaveBlk, TPB, 0, stream>>>(ei, h, dis, cat, 0);
    k_relu_slice    <<<featBlk, TPB, 0, stream>>>(cat, 0);

    // Layer 2: h = x1 @ W2 ; cat[:,64:128] = relu(agg(h) + b2)
    gemm(cat + 0, NCAT, W2, h, NHID, 0, N_NODES, NHID, NHID, nullptr);
    k_self_loop_bias<<<featBlk, TPB, 0, stream>>>(h, dis, b2, cat, 64);
    k_edge_agg      <<<edgeWaveBlk, TPB, 0, stream>>>(ei, h, dis, cat, 64);
    k_relu_slice    <<<featBlk, TPB, 0, stream>>>(cat, 64);

    // Layer 3: h = x2 @ W3 ; cat[:,128:192] = agg(h) + b3   (no relu)
    gemm(cat + 64, NCAT, W3, h, NOUT, 0, N_NODES, NOUT, NHID, nullptr);
    k_self_loop_bias<<<featBlk, TPB, 0, stream>>>(h, dis, b3, cat, 128);
    k_edge_agg      <<<edgeWaveBlk, TPB, 0, stream>>>(ei, h, dis, cat, 128);

    // Head: out = log_softmax(cat @ Wl + bl)
    gemm(cat, NCAT, Wl, out, NCLASS, 0, N_NODES, NCLASS, NCAT, bl);
    k_log_softmax<<<nodeBlk, TPB, 0, stream>>>(out);
}